// OrthonormalTransform_73856257622205
// MI455X (gfx1250) — compile-verified
//
#include <hip/hip_runtime.h>
#include <math.h>
#include <stdint.h>

// MI455X / gfx1250, wave32 only.
typedef __attribute__((ext_vector_type(2))) float v2f;
typedef __attribute__((ext_vector_type(8))) float v8f;

#define DIM_N   128
#define N_ANG   8128
#define N_COLS  524288
#define TILES_PER_BLOCK 8           // 8 * 16 = 128 columns of X per block

#define ROW_BYTES (N_COLS * 4u)     // 2 MB: byte stride between rows of X/out

// ---------------------------------------------------------------------------
// Kernel 1: build Rs = diag(mus) * (G_8127 ... G_1 G_0)  (128x128, row-major)
//
// One block, 128 threads; thread j owns column j of R (rotations mix rows
// p,q of every column independently -> no inter-thread dependency at all).
// Per inner iteration the chained value rp stays in a register, so the
// critical path is 2 VALU ops, not an LDS round trip. cos/sin are
// precomputed with precise sincosf into LDS (parallel across threads).
// Row p is final after outer iteration p -> streamed to global immediately.
// ---------------------------------------------------------------------------
__global__ void build_R_kernel(const float* __restrict__ angles,
                               const float* __restrict__ mus,
                               float* __restrict__ Rout) {
    extern __shared__ float smem[];
    float* Rl = smem;                 // DIM_N * DIM_N floats (64 KB)
    float* CS = smem + DIM_N * DIM_N; // 2 * N_ANG floats (63.5 KB)

    const int j = threadIdx.x;        // column this thread owns

    // Parallel precompute of (cos, sin) for all 8128 angles (precise).
    for (int k = j; k < N_ANG; k += DIM_N) {
        float s, c;
        sincosf(angles[k], &s, &c);
        CS[2 * k]     = c;
        CS[2 * k + 1] = s;
    }
    // Identity column (column-private, no sync needed for Rl itself).
    for (int i = 0; i < DIM_N; ++i)
        Rl[i * DIM_N + j] = (i == j) ? 1.0f : 0.0f;
    __syncthreads();   // CS must be complete

    int k = 0;
    for (int p = 0; p < DIM_N - 1; ++p) {
        float rp = Rl[p * DIM_N + j];           // row p lives in a register
        for (int q = p + 1; q < DIM_N; ++q, ++k) {
            const float c  = CS[2 * k];
            const float s  = CS[2 * k + 1];
            const float rq = Rl[q * DIM_N + j];
            Rl[q * DIM_N + j] = fmaf(s, rp, c * rq);   // uses OLD rp (matches ref)
            rp = fmaf(c, rp, -(s * rq));
        }
        // Row p is never touched by later rotations (they act on rows > p).
        Rout[p * DIM_N + j] = mus[p] * rp;
    }
    Rout[(DIM_N - 1) * DIM_N + j] = mus[DIM_N - 1] * Rl[(DIM_N - 1) * DIM_N + j];
}

// ---------------------------------------------------------------------------
// Kernel 2: out = Rs @ X via V_WMMA_F32_16X16X4_F32.
//
// Block = 256 threads = 8 waves; wave w owns the 16-row slab [16w, 16w+16) of
// Rs, held entirely in registers as 32 A-fragments (K split into 32 chunks of
// 4). Each block covers 128 consecutive columns of X as 8 column-tiles of 16;
// per tile a wave chains 32 WMMAs (accumulators alternate for ILP) and stores
// its 16x16 fp32 result. Rs (64 KB) is L2-resident; X-tile reuse across the
// 8 waves is served by WGP$/L2, so HBM traffic ~= read X + write out (512 MB
// -> ~22us floor at 23.3 TB/s).
//
// Addressing: all X/out accesses use a 32-bit byte offset off a uniform
// 64-bit base (GVS mode: saddr + 32-bit voffset + imm). The K loop advances
// 4 rows (8 MB) per step so both B loads sit at imm offsets {0, +2 MB} and
// the only per-iteration address work is one 32-bit add.
//
// WMMA f32 16x16x4 fragment mapping (CDNA5 ISA):
//   A (16x4):  lane<16: M=lane,    v.x=K0, v.y=K1 ; lane>=16: M=lane-16, K2/K3
//   B (4x16):  lane<16: N=lane,    v.x=K0, v.y=K1 ; lane>=16: N=lane-16, K2/K3
//   D (16x16): VGPR r -> rows r (lanes 0-15) and r+8 (lanes 16-31)
// ---------------------------------------------------------------------------
__global__ __launch_bounds__(256) void rot_gemm_kernel(
        const float* __restrict__ X,
        const float* __restrict__ Rs,
        float* __restrict__ out) {
    const int lane = threadIdx.x & 31;
    const int wave = threadIdx.x >> 5;     // M-tile index 0..7
    const int half = lane >> 4;            // 0: K0/K1 side, 1: K2/K3 side
    const int l16  = lane & 15;

    // A fragments: whole 16x128 slab of Rs in 64 VGPRs (one b64 load each).
    v2f a[32];
    {
        const float* rp = Rs + (wave * 16 + l16) * DIM_N + half * 2;
        #pragma unroll
        for (int k = 0; k < 32; ++k)
            a[k] = *(const v2f*)(rp + 4 * k);
    }

    const char* Xb  = (const char*)X;
    char*       Ob  = (char*)out;
    const int colBase = blockIdx.x * (16 * TILES_PER_BLOCK);

    for (int t = 0; t < TILES_PER_BLOCK; ++t) {
        const uint32_t col0 = (uint32_t)(colBase + t * 16);

        // Lane's 32-bit byte offset into X: row = 2*half, col = col0 + l16.
        uint32_t xoff = (uint32_t)(half * 2) * ROW_BYTES
                      + (col0 + (uint32_t)l16) * 4u;

        // Prefetch next tile's leading cachelines (global_prefetch_b8).
        if (t + 1 < TILES_PER_BLOCK)
            __builtin_prefetch(Xb + xoff + 64u, 0, 3);

        v8f acc0 = {};
        v8f acc1 = {};
        #pragma unroll
        for (int k = 0; k < 32; ++k) {
            v2f b;
            b.x = *(const float*)(Xb + xoff);             // row 4k   + 2*half
            b.y = *(const float*)(Xb + xoff + ROW_BYTES); // row 4k+1 + 2*half
            if (k & 1)
                acc1 = __builtin_amdgcn_wmma_f32_16x16x4_f32(
                    false, a[k], false, b, (short)0, acc1, false, false);
            else
                acc0 = __builtin_amdgcn_wmma_f32_16x16x4_f32(
                    false, a[k], false, b, (short)0, acc0, false, false);
            xoff += 4u * ROW_BYTES;   // advance 4 rows (8 MB), one 32-bit add
        }

        // Store 16x16 tile: rows (16*wave + 8*half + r), r = 0..7.
        uint32_t ooff = (uint32_t)(wave * 16 + half * 8) * ROW_BYTES
                      + (col0 + (uint32_t)l16) * 4u;
        #pragma unroll
        for (int r = 0; r < 8; ++r)
            *(float*)(Ob + ooff + (uint32_t)r * ROW_BYTES) = acc0[r] + acc1[r];
    }
}

// ---------------------------------------------------------------------------
extern "C" void kernel_launch(void* const* d_in, const int* in_sizes, int n_in,
                              void* d_out, int out_size, void* d_ws, size_t ws_size,
                              hipStream_t stream) {
    const float* X      = (const float*)d_in[0];   // (128, 524288) fp32
    const float* angles = (const float*)d_in[1];   // (8128,) fp32
    const float* mus    = (const float*)d_in[2];   // (128,) fp32
    float* out = (float*)d_out;                    // (128, 524288) fp32
    float* Rs  = (float*)d_ws;                     // 128*128 fp32 = 64 KB

    const size_t lds_bytes = (size_t)(DIM_N * DIM_N + 2 * N_ANG) * sizeof(float); // ~127.5 KB
    // CDNA5 allows up to 320 KB LDS per workgroup; raise the dynamic-LDS cap.
    (void)hipFuncSetAttribute((const void*)build_R_kernel,
                              hipFuncAttributeMaxDynamicSharedMemorySize,
                              (int)lds_bytes);

    build_R_kernel<<<1, DIM_N, lds_bytes, stream>>>(angles, mus, Rs);

    const int grid = N_COLS / (16 * TILES_PER_BLOCK);  // 4096 blocks
    rot_gemm_kernel<<<grid, 256, 0, stream>>>(X, Rs, out);
}